// ConvEmbedding_850403525141
// MI455X (gfx1250) — compile-verified
//
#include <hip/hip_runtime.h>
#include <hip/hip_bf16.h>

// Problem constants (from reference): B=16, N=2048, C=2, H=128, K+1=10
#define NB   16
#define NPTS 2048
#define HDIM 128
#define KW   10      // kernel window (K+1)
#define GQ   20      // gathered vector length = KW * C

typedef float v2f __attribute__((ext_vector_type(2)));
typedef float v8f __attribute__((ext_vector_type(8)));

// ---------------------------------------------------------------------------
// Kernel 0: fold conv weights through W2, store PAIR-INTERLEAVED for b64 loads:
//   Wbig[q][e]  = sum_h Wconv[h][c][k] * W2[e][h],  q = k*2 + c   (20 x 128)
//   Wpack[p][e][r] = Wbig[2p + r][e],  p = 0..9, r = 0..1
//   bbig[e]     = b2[e] + sum_h bconv[h] * W2[e][h]
// One block, 128 threads (thread e).
// ---------------------------------------------------------------------------
__global__ __launch_bounds__(128)
void prep_weights_kernel(const float* __restrict__ Wconv,
                         const float* __restrict__ bconv,
                         const float* __restrict__ W2,
                         const float* __restrict__ b2,
                         float* __restrict__ Wpack,
                         float* __restrict__ bbig) {
    const int e = threadIdx.x;   // 0..127
    float acc[GQ];
#pragma unroll
    for (int q = 0; q < GQ; ++q) acc[q] = 0.0f;
    float bacc = b2[e];
    for (int h = 0; h < HDIM; ++h) {
        const float w2 = W2[e * HDIM + h];
        bacc += bconv[h] * w2;
        const float* wc = Wconv + h * (2 * KW);   // Wconv[h][c][k], c-major rows of 10
#pragma unroll
        for (int k = 0; k < KW; ++k) {
            acc[k * 2 + 0] += wc[k]      * w2;    // c = 0
            acc[k * 2 + 1] += wc[KW + k] * w2;    // c = 1
        }
    }
#pragma unroll
    for (int q = 0; q < GQ; ++q)
        Wpack[(q >> 1) * (HDIM * 2) + e * 2 + (q & 1)] = acc[q];
    bbig[e] = bacc;
}

// ---------------------------------------------------------------------------
// Kernel 1: per-point KNN (top-10 smallest distances, reference tie order)
// and gather into G (B*N x 20), G[m][(9-j)*2+c] = x[asc[j]][c]
// (ascending list flipped -> reference's descending window, self/nearest last).
// Grid: (N/256, B); block 256 threads; whole batch x + sq in LDS (24 KB).
// ---------------------------------------------------------------------------
__global__ __launch_bounds__(256)
void knn_gather_kernel(const float* __restrict__ x, float* __restrict__ G) {
    __shared__ float xs0[NPTS];
    __shared__ float xs1[NPTS];
    __shared__ float sq[NPTS];

    const int b   = blockIdx.y;
    const int tid = threadIdx.x;
    const float* xb = x + (size_t)b * NPTS * 2;

    for (int i = tid; i < NPTS; i += 256) {
        const float a0 = xb[i * 2 + 0];
        const float a1 = xb[i * 2 + 1];
        xs0[i] = a0;
        xs1[i] = a1;
        sq[i]  = a0 * a0 + a1 * a1;
    }
    __syncthreads();

    const int n  = blockIdx.x * 256 + tid;
    const float q0  = xs0[n];
    const float q1  = xs1[n];
    const float qsq = sq[n];

    float bd[KW];
    int   bi[KW];
#pragma unroll
    for (int j = 0; j < KW; ++j) { bd[j] = 3.4e38f; bi[j] = 0; }

    for (int m = 0; m < NPTS; ++m) {
        // Match reference formula: sq_n + sq_m - 2*dot, clamped at 0.
        float d2 = (qsq + sq[m]) - 2.0f * (q0 * xs0[m] + q1 * xs1[m]);
        d2 = fmaxf(d2, 0.0f);
        if (d2 < bd[KW - 1]) {          // strict < keeps stable (argsort) tie order
            bd[KW - 1] = d2;
            bi[KW - 1] = m;
#pragma unroll
            for (int j = KW - 1; j > 0; --j) {
                if (bd[j] < bd[j - 1]) {
                    const float td = bd[j]; bd[j] = bd[j - 1]; bd[j - 1] = td;
                    const int   ti = bi[j]; bi[j] = bi[j - 1]; bi[j - 1] = ti;
                }
            }
        }
    }

    const size_t row = (size_t)b * NPTS + n;
    float* g = G + row * GQ;
#pragma unroll
    for (int j = 0; j < KW; ++j) {
        const int idx = bi[j];
        g[(KW - 1 - j) * 2 + 0] = xs0[idx];
        g[(KW - 1 - j) * 2 + 1] = xs1[idx];
    }
}

// ---------------------------------------------------------------------------
// Kernel 2: Out = G(32768x20) @ Wbig(20x128) + bbig + b1 + x@W1^T  (fp32 WMMA)
// 8 waves/block, one 16-row tile per wave, 8 column tiles of 16, K=20 = 5
// v_wmma_f32_16x16x4_f32 steps per tile.
// A layout (ISA 7.12.2): lanes 0-15 M=lane, K=0/1 in v0/v1; lanes 16-31 K=2/3.
// B layout mirrors A (per 8/16-bit B patterns): lane-half h holds K rows
//   krow = s*4 + h*2 + {0,1} for column l16 -> one b64 load from Wpack.
// C/D layout: VGPR r -> M = r (lanes 0-15) / r+8 (lanes 16-31), col = lane&15.
// ---------------------------------------------------------------------------
__global__ __launch_bounds__(256)
void gemm_wmma_kernel(const float* __restrict__ G,
                      const float* __restrict__ Wpack,
                      const float* __restrict__ bbig,
                      const float* __restrict__ x,
                      const float* __restrict__ W1,
                      const float* __restrict__ b1,
                      float* __restrict__ out) {
    const int wave   = threadIdx.x >> 5;          // 0..7
    const int lane   = threadIdx.x & 31;
    const int lhalf  = lane >> 4;                 // 0 or 1
    const int l16    = lane & 15;
    const int m0     = (blockIdx.x * 8 + wave) * 16;   // row-tile base

    // A fragments for all 5 K-steps (row = m0 + l16, K = s*4 + lhalf*2 + {0,1})
    v2f afrag[5];
    const float* grow = G + (size_t)(m0 + l16) * GQ + lhalf * 2;
#pragma unroll
    for (int s = 0; s < 5; ++s) {
        afrag[s] = *(const v2f*)(grow + s * 4);
    }

    // Epilogue row data: rows m = m0 + lhalf*8 + r
    float xr0[8], xr1[8];
#pragma unroll
    for (int r = 0; r < 8; ++r) {
        const int m = m0 + lhalf * 8 + r;
        const v2f xv = *(const v2f*)(x + (size_t)m * 2);
        xr0[r] = xv.x;
        xr1[r] = xv.y;
    }

    for (int ct = 0; ct < 8; ++ct) {              // column tiles of 16
        const int e = ct * 16 + l16;              // this lane's output column
        v8f c = {};
#pragma unroll
        for (int s = 0; s < 5; ++s) {
            // B fragment pair index p = krow/2 = s*2 + lhalf; single b64 load.
            const v2f bfrag =
                *(const v2f*)(Wpack + (s * 2 + lhalf) * (HDIM * 2) + e * 2);
            c = __builtin_amdgcn_wmma_f32_16x16x4_f32(
                    /*neg_a=*/false, afrag[s],
                    /*neg_b=*/false, bfrag,
                    /*c_mod=*/(short)0, c,
                    /*reuse_a=*/false, /*reuse_b=*/false);
        }
        const v2f w1v  = *(const v2f*)(W1 + e * 2);
        const float bias = bbig[e] + b1[e];
#pragma unroll
        for (int r = 0; r < 8; ++r) {
            const int m = m0 + lhalf * 8 + r;
            out[(size_t)m * HDIM + e] = c[r] + bias + xr0[r] * w1v.x + xr1[r] * w1v.y;
        }
    }
}

// ---------------------------------------------------------------------------
extern "C" void kernel_launch(void* const* d_in, const int* in_sizes, int n_in,
                              void* d_out, int out_size, void* d_ws, size_t ws_size,
                              hipStream_t stream) {
    const float* x     = (const float*)d_in[0];
    const float* Wconv = (const float*)d_in[1];
    const float* bconv = (const float*)d_in[2];
    const float* W1    = (const float*)d_in[3];
    const float* b1    = (const float*)d_in[4];
    const float* W2    = (const float*)d_in[5];
    const float* b2    = (const float*)d_in[6];
    float* out = (float*)d_out;

    // Workspace layout (fp32): G (32768 x 20), Wpack (10 x 128 x 2), bbig (128)
    float* G     = (float*)d_ws;
    float* Wpack = G + (size_t)NB * NPTS * GQ;
    float* bbig  = Wpack + GQ * HDIM;

    prep_weights_kernel<<<1, 128, 0, stream>>>(Wconv, bconv, W2, b2, Wpack, bbig);

    dim3 g1(NPTS / 256, NB);
    knn_gather_kernel<<<g1, 256, 0, stream>>>(x, G);

    // 32768 rows / (8 waves * 16 rows) = 256 blocks
    gemm_wmma_kernel<<<256, 256, 0, stream>>>(G, Wpack, bbig, x, W1, b1, out);
}